// StabilizerEmbedder_27719718928452
// MI455X (gfx1250) — compile-verified
//
#include <hip/hip_runtime.h>

typedef __attribute__((ext_vector_type(2))) float v2f;
typedef __attribute__((ext_vector_type(4))) float v4f;
typedef __attribute__((ext_vector_type(8))) float v8f;

// Problem dims (fixed by the reference)
constexpr int kB = 128;
constexpr int kS = 1024;
constexpr int kF = 16;
constexpr int kH = 512;
constexpr int kM = kB * kS;          // 131072 flattened rows
constexpr int kRowsPerBlock = 16;    // one 16-row stripe per block
constexpr int kLdsW = 516;           // padded LDS row stride (floats): kills bank conflicts
constexpr float kInvH = 1.0f / (float)kH;
constexpr float kLnEps = 1e-5f;

__global__ __launch_bounds__(512)
void stab_embed_ln_fused(const float* __restrict__ x,
                         const float* __restrict__ W,
                         const float* __restrict__ bias,
                         const float* __restrict__ emb,
                         const float* __restrict__ gamma,
                         const float* __restrict__ beta,
                         float* __restrict__ out)
{
    __shared__ float lds_h[kRowsPerBlock * kLdsW];  // 33 KB h-tile
    __shared__ float lds_bsum[kH];                  // 2 KB column bias sums

    const int tid  = threadIdx.x;
    const int wave = tid >> 5;        // 16 waves
    const int lane = tid & 31;
    const int m0   = blockIdx.x * kRowsPerBlock;

    // ---- per-column bias sums (one column per thread) ----
    {
        float s = 0.0f;
        #pragma unroll
        for (int f = 0; f < kF; ++f) s += bias[f * kH + tid];
        lds_bsum[tid] = s;
    }

    // ---- GEMM (M=16,N=512,K=16) via chained v_wmma_f32_16x16x4_f32 ----
    // A-matrix 16x4 f32 layout: lanes 0-15 hold K={k0,k0+1}, lanes 16-31 hold K={k0+2,k0+3}
    const int mrow = lane & 15;        // A row held by this lane
    const int koff = (lane >> 4) * 2;  // K sub-offset for this half-wave
    const int nl   = lane & 15;        // B/C/D column within a 16-wide tile

    v2f afrag[4];
    #pragma unroll
    for (int kk = 0; kk < 4; ++kk) {
        const float* ap = x + (size_t)(m0 + mrow) * kF + kk * 4 + koff;
        afrag[kk] = *(const v2f*)ap;   // 8B-aligned global_load_b64
    }

    #pragma unroll
    for (int t = 0; t < 2; ++t) {      // each wave: two 16-col tiles -> 32 tiles = 512 cols
        const int n0 = (wave * 2 + t) * 16;
        v8f c = {};
        #pragma unroll
        for (int kk = 0; kk < 4; ++kk) {
            const int k = kk * 4 + koff;
            v2f bfrag;
            bfrag.x = W[(k + 0) * kH + n0 + nl];
            bfrag.y = W[(k + 1) * kH + n0 + nl];
            c = __builtin_amdgcn_wmma_f32_16x16x4_f32(
                    false, afrag[kk], false, bfrag, (short)0, c, false, false);
        }
        // C/D layout: VGPR r -> row r (lanes 0-15) / row r+8 (lanes 16-31), col = lane&15
        const int rbase = (lane >> 4) * 8;
        #pragma unroll
        for (int r = 0; r < 8; ++r)
            lds_h[(rbase + r) * kLdsW + n0 + nl] = c[r];
    }

    __syncthreads();

    // ---- bias + embedding + LayerNorm: one wave per row ----
    const int row  = m0 + wave;        // global flattened row
    const int srow = row % kS;         // stabilizer index (rows are b*S + s)

    float vbuf[16];
    float sum = 0.0f, sumsq = 0.0f;
    #pragma unroll
    for (int j = 0; j < 4; ++j) {
        const int col = (lane + j * 32) * 4;   // coalesced float4 chunks
        v4f hv = *(const v4f*)&lds_h[wave * kLdsW + col];
        v4f bs = *(const v4f*)&lds_bsum[col];
        v4f ev = *(const v4f*)&emb[(size_t)srow * kH + col];
        #pragma unroll
        for (int q = 0; q < 4; ++q) {
            float v = hv[q] + bs[q] + ev[q];
            vbuf[j * 4 + q] = v;
            sum   += v;
            sumsq += v * v;
        }
    }

    // wave32 butterfly reduction
    #pragma unroll
    for (int off = 16; off >= 1; off >>= 1) {
        sum   += __shfl_xor(sum,   off, 32);
        sumsq += __shfl_xor(sumsq, off, 32);
    }
    const float mean = sum * kInvH;
    const float var  = sumsq * kInvH - mean * mean;
    const float rstd = rsqrtf(var + kLnEps);

    #pragma unroll
    for (int j = 0; j < 4; ++j) {
        const int col = (lane + j * 32) * 4;
        v4f g  = *(const v4f*)&gamma[col];
        v4f be = *(const v4f*)&beta[col];
        v4f o;
        #pragma unroll
        for (int q = 0; q < 4; ++q)
            o[q] = (vbuf[j * 4 + q] - mean) * rstd * g[q] + be[q];
        *(v4f*)&out[(size_t)row * kH + col] = o;   // global_store_b128, fully coalesced
    }
}

extern "C" void kernel_launch(void* const* d_in, const int* in_sizes, int n_in,
                              void* d_out, int out_size, void* d_ws, size_t ws_size,
                              hipStream_t stream) {
    (void)in_sizes; (void)n_in; (void)d_ws; (void)ws_size; (void)out_size;
    const float* x     = (const float*)d_in[0];
    const float* W     = (const float*)d_in[1];
    const float* bias  = (const float*)d_in[2];
    const float* emb   = (const float*)d_in[3];
    const float* gamma = (const float*)d_in[4];
    const float* beta  = (const float*)d_in[5];
    float* out = (float*)d_out;

    dim3 grid(kM / kRowsPerBlock);   // 8192 blocks
    dim3 block(512);                 // 16 wave32s
    hipLaunchKernelGGL(stab_embed_ln_fused, grid, block, 0, stream,
                       x, W, bias, emb, gamma, beta, out);
}